// NaiveNeuralAdaptiveBias_82291573391763
// MI455X (gfx1250) — compile-verified
//
#include <hip/hip_runtime.h>
#include <hip/hip_bf16.h>
#include <math.h>

typedef __attribute__((ext_vector_type(2))) float v2f;
typedef __attribute__((ext_vector_type(8))) float v8f;

#define BATCH 8
#define NPTS  512
#define EMB   128

// One wave32 handles 16 consecutive flat (b,i,j) pairs.
// Layer 1: 8x V_WMMA_F32_16X16X4_F32 with bias folded into the K=3 slot:
//   A = [angle, cost, dur, 1.0] (16x4), B = [W1_row0; W1_row1; W1_row2; b1] (4x16 chunk)
//   => C starts at zero, which lowers to the inline-0 SRC2 WMMA form.
// Layer 2: in-register silu (fast rcp) + dot with W2 + shfl_xor reduction.
__global__ __launch_bounds__(256) void nab_kernel(
    const float* __restrict__ coords,   // [B,N,2]
    const float* __restrict__ cost,     // [B,N,N]
    const float* __restrict__ dur,      // [B,N,N]
    const float* __restrict__ W1,       // [3,EMB]
    const float* __restrict__ b1,       // [EMB]
    const float* __restrict__ W2,       // [EMB,1]
    const float* __restrict__ b2,       // [1]
    float* __restrict__ out)            // [B,N,N]
{
    const int lane     = threadIdx.x & 31;
    const int waveId   = (blockIdx.x * blockDim.x + threadIdx.x) >> 5;
    const int pairBase = waveId << 4;          // 16 pairs per wave
    const int n        = lane & 15;            // column / row-in-tile index
    const bool hiHalf  = lane >= 16;

    // ---- A fragment (16x4 f32 layout):
    //  lanes 0-15 : v0 = A[n][0] = angle , v1 = A[n][1] = cost
    //  lanes 16-31: v0 = A[n][2] = dur   , v1 = A[n][3] = 1.0 (bias row selector)
    const int p   = pairBase + n;
    const int b   = p >> 18;                   // / (512*512)
    const int rem = p & 0x3FFFF;
    const int i   = rem >> 9;
    const int j   = rem & 511;

    v2f a;
    if (!hiHalf) {
        const float xi = coords[(b * NPTS + i) * 2 + 0];
        const float yi = coords[(b * NPTS + i) * 2 + 1];
        const float xj = coords[(b * NPTS + j) * 2 + 0];
        const float yj = coords[(b * NPTS + j) * 2 + 1];
        a.x = atan2f(yi - yj, xi - xj);
        a.y = cost[p];
    } else {
        a.x = dur[p];
        a.y = 1.0f;
    }

    // ---- B fragments (4x16 per 16-wide embed chunk); K=3 row carries b1
    v2f bw[8];
#pragma unroll
    for (int c = 0; c < 8; ++c) {
        const int e = c * 16 + n;
        if (!hiHalf) {
            bw[c].x = W1[0 * EMB + e];         // K=0 row
            bw[c].y = W1[1 * EMB + e];         // K=1 row
        } else {
            bw[c].x = W1[2 * EMB + e];         // K=2 row
            bw[c].y = b1[e];                   // K=3 row = bias
        }
    }

    // ---- Layer 1: D = A x B + 0  (8 chunks cover EMB=128)
    v8f acc[8];
#pragma unroll
    for (int c = 0; c < 8; ++c) {
        v8f zero = {};
        acc[c] = __builtin_amdgcn_wmma_f32_16x16x4_f32(
            false, a, false, bw[c], (short)0, zero, false, false);
    }

    // ---- SiLU + layer-2 partial dot products (per-lane, 64 elems)
    // silu(x) = x * rcp(1 + exp(-x)); denominator >= 1, so raw v_rcp_f32 is safe.
    float partial[8];
#pragma unroll
    for (int v = 0; v < 8; ++v) partial[v] = 0.0f;
#pragma unroll
    for (int c = 0; c < 8; ++c) {
        const float w2c = W2[c * 16 + n];
#pragma unroll
        for (int v = 0; v < 8; ++v) {
            const float x = acc[c][v];
            const float s = x * __builtin_amdgcn_rcpf(1.0f + __expf(-x));
            partial[v] = fmaf(s, w2c, partial[v]);
        }
    }

    // ---- reduce across the 16 lanes of each half (xor masks stay in-group)
#pragma unroll
    for (int v = 0; v < 8; ++v) {
        partial[v] += __shfl_xor(partial[v], 8, 32);
        partial[v] += __shfl_xor(partial[v], 4, 32);
        partial[v] += __shfl_xor(partial[v], 2, 32);
        partial[v] += __shfl_xor(partial[v], 1, 32);
    }

    // ---- lanes n<8 of each half write rows m = n (+8 for hi half)
    float r = partial[0];
    r = (n == 1) ? partial[1] : r;
    r = (n == 2) ? partial[2] : r;
    r = (n == 3) ? partial[3] : r;
    r = (n == 4) ? partial[4] : r;
    r = (n == 5) ? partial[5] : r;
    r = (n == 6) ? partial[6] : r;
    r = (n == 7) ? partial[7] : r;
    if (n < 8) {
        const int m = n + (hiHalf ? 8 : 0);
        out[pairBase + m] = r + b2[0];
    }
}

extern "C" void kernel_launch(void* const* d_in, const int* in_sizes, int n_in,
                              void* d_out, int out_size, void* d_ws, size_t ws_size,
                              hipStream_t stream) {
    const float* coords = (const float*)d_in[0];
    const float* cost   = (const float*)d_in[1];
    const float* dur    = (const float*)d_in[2];
    const float* W1     = (const float*)d_in[3];
    const float* b1     = (const float*)d_in[4];
    const float* W2     = (const float*)d_in[5];
    const float* b2     = (const float*)d_in[6];
    float* out = (float*)d_out;

    const int totalPairs = BATCH * NPTS * NPTS;     // 2,097,152
    const int waves      = totalPairs / 16;         // 131,072
    const int threads    = 256;                     // 8 waves/block
    const int blocks     = waves / (threads / 32);  // 16,384

    hipLaunchKernelGGL(nab_kernel, dim3(blocks), dim3(threads), 0, stream,
                       coords, cost, dur, W1, b1, W2, b2, out);
}